// Edge_Embedding_Model_22660247454116
// MI455X (gfx1250) — compile-verified
//
#include <hip/hip_runtime.h>
#include <cstdint>
#include <cstddef>

typedef float v2f __attribute__((ext_vector_type(2)));
typedef float v8f __attribute__((ext_vector_type(8)));

#define DIM 128
#define KGROUPS 32          // 128 / 4  (K per V_WMMA_F32_16X16X4_F32)
#define NBLK 8              // 128 / 16 (output column blocks)
#define WSTRIDE 132         // padded LDS row stride in dwords: conflict-free b64 reads
#define WMAT_DW (DIM * WSTRIDE)   // one 128x128 weight matrix staged in LDS (dwords)
#define SCR_DW (16 * WSTRIDE)     // per-wave 16x128 scratch tile (dwords)
#define BTHREADS 512
#define BWAVES 16

__device__ __forceinline__ v8f wmma_f32_k4(v2f a, v2f b, v8f c) {
  // D(16x16,f32) = A(16x4,f32) * B(4x16,f32) + C
  return __builtin_amdgcn_wmma_f32_16x16x4_f32(false, a, false, b, (short)0, c,
                                               false, false);
}

// One 16xDIM GEMM tile: acc += A_tile @ W^T, W staged in LDS at `lw`.
// Batched B-fragment loads per K-group so DS latency is amortized over 8 WMMAs.
__device__ __forceinline__ void gemm_tile(const float* __restrict__ Arow,
                                          const float* __restrict__ lw,
                                          int r, int koff, v8f acc[NBLK]) {
  #pragma unroll 2
  for (int kg = 0; kg < KGROUPS; ++kg) {
    const v2f a = *(const v2f*)(Arow + kg * 4 + koff);
    v2f b[NBLK];
    #pragma unroll
    for (int nb = 0; nb < NBLK; ++nb)
      b[nb] = *(const v2f*)(lw + (nb * 16 + r) * WSTRIDE + kg * 4 + koff);
    #pragma unroll
    for (int nb = 0; nb < NBLK; ++nb)
      acc[nb] = wmma_f32_k4(a, b[nb], acc[nb]);
  }
}

__device__ __forceinline__ void store_tile(float* __restrict__ out, int row0,
                                           int limit, int r, int h,
                                           const v8f acc[NBLK]) {
  if (row0 + 16 <= limit) {                 // full tile: unguarded stores
    #pragma unroll
    for (int nb = 0; nb < NBLK; ++nb)
      #pragma unroll
      for (int rr = 0; rr < 8; ++rr)
        out[(size_t)(row0 + rr + 8 * h) * DIM + nb * 16 + r] = acc[nb][rr];
  } else {                                   // tail tile
    #pragma unroll
    for (int nb = 0; nb < NBLK; ++nb)
      #pragma unroll
      for (int rr = 0; rr < 8; ++rr) {
        const int m = row0 + rr + 8 * h;
        if (m < limit) out[(size_t)m * DIM + nb * 16 + r] = acc[nb][rr];
      }
  }
}

// ---------------------------------------------------------------------------
// Kernel A: src_embed = x @ W_src^T ; rx_embed = x @ W_rx^T
// 16 waves/block, one 16-row tile each; 135KB LDS -> 2 workgroups per WGP.
// ---------------------------------------------------------------------------
__global__ void __launch_bounds__(BTHREADS, 2)
node_embed_kernel(const float* __restrict__ x,
                  const float* __restrict__ Wsrc,
                  const float* __restrict__ Wrx,
                  float* __restrict__ out_src,
                  float* __restrict__ out_rx,
                  int N)
{
  extern __shared__ float smem[];

  const int tid = threadIdx.x;
  for (int i = tid; i < 2 * DIM * DIM; i += BTHREADS) {
    const int w   = i >> 14;
    const int rem = i & (DIM * DIM - 1);
    smem[w * WMAT_DW + (rem >> 7) * WSTRIDE + (rem & (DIM - 1))] =
        (w ? Wrx : Wsrc)[rem];
  }
  __syncthreads();

  const int lane = tid & 31;
  const int wave = tid >> 5;
  const int r    = lane & 15;
  const int h    = lane >> 4;
  const int koff = 2 * h;

  const int ntiles = (N + 15) >> 4;
  const int tile   = blockIdx.x * BWAVES + wave;
  if (tile >= ntiles) return;               // wave-uniform

  const int row0 = tile << 4;
  int arow = row0 + r;
  if (arow >= N) arow = N - 1;              // clamp loads, keep EXEC all-ones
  const float* Arow = x + (size_t)arow * DIM;

  #pragma unroll
  for (int w = 0; w < 2; ++w) {
    v8f acc[NBLK] = {};
    gemm_tile(Arow, smem + w * WMAT_DW, r, koff, acc);
    store_tile(w ? out_rx : out_src, row0, N, r, h, acc);
  }
}

// ---------------------------------------------------------------------------
// Kernel B: edge_embed = edge_attr @ W_edge^T (stored);
//           updated_edge = (edge_embed + src_embed[u] + rx_embed[v]) @ W_agg^T
// edge_embed tile parked in per-wave LDS scratch so agg A-fragments can be
// rebuilt in WMMA A-layout without cross-lane shuffles; gathers hit the 192MB
// L2 (both node-embedding arrays are L2-resident).
// ---------------------------------------------------------------------------
__global__ void __launch_bounds__(BTHREADS, 1)
edge_update_kernel(const float* __restrict__ edge_attr,
                   const long long* __restrict__ ei,   // [2, E] int64
                   const float* __restrict__ src_embed,
                   const float* __restrict__ rx_embed,
                   const float* __restrict__ Wedge,
                   const float* __restrict__ Wagg,
                   float* __restrict__ out_edge,
                   float* __restrict__ out_upd,
                   int E)
{
  extern __shared__ float smem[];

  const int tid = threadIdx.x;
  for (int i = tid; i < 2 * DIM * DIM; i += BTHREADS) {
    const int w   = i >> 14;
    const int rem = i & (DIM * DIM - 1);
    smem[w * WMAT_DW + (rem >> 7) * WSTRIDE + (rem & (DIM - 1))] =
        (w ? Wagg : Wedge)[rem];
  }
  __syncthreads();

  const int lane = tid & 31;
  const int wave = tid >> 5;
  const int r    = lane & 15;
  const int h    = lane >> 4;
  const int koff = 2 * h;

  const int etiles = (E + 15) >> 4;
  const int tile   = blockIdx.x * BWAVES + wave;
  if (tile >= etiles) return;

  const int row0 = tile << 4;
  int e = row0 + r;
  if (e >= E) e = E - 1;
  const float* Arow = edge_attr + (size_t)e * DIM;

  float* wscr = smem + 2 * WMAT_DW + wave * SCR_DW;   // private 16x128 tile

  // ---- GEMM 1: edge_embed tile -----------------------------------------
  {
    v8f acc[NBLK] = {};
    gemm_tile(Arow, smem, r, koff, acc);
    // park in LDS scratch (row-major, padded) for A-layout reload
    #pragma unroll
    for (int nb = 0; nb < NBLK; ++nb)
      #pragma unroll
      for (int rr = 0; rr < 8; ++rr)
        wscr[(rr + 8 * h) * WSTRIDE + nb * 16 + r] = acc[nb][rr];
    store_tile(out_edge, row0, E, r, h, acc);
  }

  // ---- gather rows for this wave's 16 edges ----------------------------
  const long long ue = ei[e];              // edge_index[0][e]
  const long long ve = ei[(size_t)E + e];  // edge_index[1][e]
  const float* srow = src_embed + (size_t)ue * DIM;
  const float* rrow = rx_embed + (size_t)ve * DIM;

  // ---- GEMM 2: updated_edge = agg @ W_agg^T ----------------------------
  {
    const float* lw = smem + WMAT_DW;      // W_agg
    v8f acc[NBLK] = {};
    #pragma unroll 2
    for (int kg = 0; kg < KGROUPS; ++kg) {
      const v2f ae = *(const v2f*)(wscr + r * WSTRIDE + kg * 4 + koff);
      const v2f as = *(const v2f*)(srow + kg * 4 + koff);
      const v2f ar = *(const v2f*)(rrow + kg * 4 + koff);
      const v2f a  = ae + as + ar;
      v2f b[NBLK];
      #pragma unroll
      for (int nb = 0; nb < NBLK; ++nb)
        b[nb] = *(const v2f*)(lw + (nb * 16 + r) * WSTRIDE + kg * 4 + koff);
      #pragma unroll
      for (int nb = 0; nb < NBLK; ++nb)
        acc[nb] = wmma_f32_k4(a, b[nb], acc[nb]);
    }
    store_tile(out_upd, row0, E, r, h, acc);
  }
}

// ---------------------------------------------------------------------------
extern "C" void kernel_launch(void* const* d_in, const int* in_sizes, int n_in,
                              void* d_out, int out_size, void* d_ws, size_t ws_size,
                              hipStream_t stream)
{
  const float*     x         = (const float*)d_in[0];
  const long long* ei        = (const long long*)d_in[1];   // int64 [2,E]
  const float*     edge_attr = (const float*)d_in[2];
  const float*     Wsrc      = (const float*)d_in[3];
  const float*     Wedge     = (const float*)d_in[4];
  const float*     Wrx       = (const float*)d_in[5];
  const float*     Wagg      = (const float*)d_in[6];

  const int N = in_sizes[0] / DIM;
  const int E = in_sizes[2] / DIM;

  float* out      = (float*)d_out;
  float* out_src  = out;
  float* out_edge = out_src + (size_t)N * DIM;
  float* out_rx   = out_edge + (size_t)E * DIM;
  float* out_upd  = out_rx + (size_t)N * DIM;

  const size_t ldsA = (size_t)2 * WMAT_DW * sizeof(float);               // 135168 B
  const size_t ldsB = ldsA + (size_t)BWAVES * SCR_DW * sizeof(float);    // 270336 B
  (void)hipFuncSetAttribute(reinterpret_cast<const void*>(node_embed_kernel),
                            hipFuncAttributeMaxDynamicSharedMemorySize, (int)ldsA);
  (void)hipFuncSetAttribute(reinterpret_cast<const void*>(edge_update_kernel),
                            hipFuncAttributeMaxDynamicSharedMemorySize, (int)ldsB);

  const int ntiles = (N + 15) / 16;
  const int ablk   = (ntiles + BWAVES - 1) / BWAVES;
  node_embed_kernel<<<ablk, BTHREADS, ldsA, stream>>>(x, Wsrc, Wrx, out_src,
                                                      out_rx, N);

  const int etiles = (E + 15) / 16;
  const int bblk   = (etiles + BWAVES - 1) / BWAVES;
  edge_update_kernel<<<bblk, BTHREADS, ldsB, stream>>>(edge_attr, ei, out_src,
                                                       out_rx, Wedge, Wagg,
                                                       out_edge, out_upd, E);
}